// CVLoss_65137474011368
// MI455X (gfx1250) — compile-verified
//
#include <hip/hip_runtime.h>
#include <stdint.h>
#include <math.h>

// ---------------- problem constants (from reference) ----------------
#define N_NEURONS 1024
#define T1        256          // stage-1 threads per block (8 wave32s)
#define SLOTS     4            // async pipeline depth (LDS ring slots)
#define ROW_BYTES (N_NEURONS * 4)   // one time-row of spikes = 4 KB

// ---------------- CDNA5 async global->LDS helpers ----------------
// GLOBAL_LOAD_ASYNC_TO_LDS_B128 (GV mode: 64-bit vaddr, saddr=off).
// Non-temporal: the spike tensor is streamed exactly once.
__device__ __forceinline__ void async_row_load_b128(unsigned lds_byte_addr,
                                                    unsigned long long gaddr) {
  asm volatile("global_load_async_to_lds_b128 %0, %1, off th:TH_LOAD_NT"
               :: "v"(lds_byte_addr), "v"(gaddr)
               : "memory");
}
#define WAIT_ASYNC_LE2() asm volatile("s_wait_asynccnt 2" ::: "memory")
#define WAIT_ASYNC_0()   asm volatile("s_wait_asynccnt 0" ::: "memory")
#define WAIT_DS_0()      asm volatile("s_wait_dscnt 0"    ::: "memory")

// Per-element ISI state update. Spikes are exactly 0.0f / 1.0f, so the count
// update is a plain add (no compare/select).
__device__ __forceinline__ void isi_update(float x, float tf,
                                           float& cnt, float& first,
                                           float& last, float& sq) {
  const bool  sp  = x > 0.f;
  const bool  had = cnt > 0.f;
  const float g   = tf - last;
  sq   += (sp && had) ? g * g : 0.f;
  first = (sp && !had) ? tf : first;
  last  = sp ? tf : last;
  cnt  += x;                       // binary spike train
}

// ---------------- stage 1: per-(chunk, neuron) ISI partial state ----------------
// Each block owns one contiguous time chunk of `chlen` rows; each thread owns
// 4 adjacent neurons (one float4 per row; a wave loads 512 contiguous bytes).
// Rows stream through a 4-slot LDS ring via async copies (ASYNCcnt), consumed
// with ds_load_b128. Output per (chunk, neuron): {count, t_first, t_last, sum_gap2}.
__global__ __launch_bounds__(T1) void cv_stage1(const float* __restrict__ spikes,
                                                float4* __restrict__ part,
                                                int chlen) {
  const int c   = blockIdx.x;
  const int tid = threadIdx.x;
  const int t0  = c * chlen;

  __shared__ float4 buf[SLOTS][T1];   // 16 KB ring

  const char* src = (const char*)spikes + (size_t)t0 * ROW_BYTES + (size_t)tid * 16;
  // Low 32 bits of a generic pointer into LDS are the LDS byte offset.
  const unsigned lds0      = (unsigned)(size_t)(&buf[0][tid]);
  const unsigned slot_step = T1 * 16;

  float cnt[4], first[4], last[4], sq[4];
#pragma unroll
  for (int j = 0; j < 4; ++j) { cnt[j] = 0.f; first[j] = -1.f; last[j] = -1.f; sq[j] = 0.f; }

  // Prologue: fill the ring (rows t0 .. t0+SLOTS-1).
#pragma unroll
  for (int s = 0; s < SLOTS; ++s)
    async_row_load_b128(lds0 + (unsigned)s * slot_step,
                        (unsigned long long)(src + (size_t)s * ROW_BYTES));

  // Main loop, unrolled x2. chlen is a multiple of 4 and >= SLOTS.
  int i = 0;
  for (; i < chlen - SLOTS; i += 2) {
    WAIT_ASYNC_LE2();                         // rows i and i+1 resident
    float4 v0 = buf[i & (SLOTS - 1)][tid];    // ds_load_b128
    float4 v1 = buf[(i + 1) & (SLOTS - 1)][tid];
    const float tf0 = (float)(t0 + i);
    const float tf1 = tf0 + 1.f;
    isi_update(v0.x, tf0, cnt[0], first[0], last[0], sq[0]);
    isi_update(v0.y, tf0, cnt[1], first[1], last[1], sq[1]);
    isi_update(v0.z, tf0, cnt[2], first[2], last[2], sq[2]);
    isi_update(v0.w, tf0, cnt[3], first[3], last[3], sq[3]);
    isi_update(v1.x, tf1, cnt[0], first[0], last[0], sq[0]);
    isi_update(v1.y, tf1, cnt[1], first[1], last[1], sq[1]);
    isi_update(v1.z, tf1, cnt[2], first[2], last[2], sq[2]);
    isi_update(v1.w, tf1, cnt[3], first[3], last[3], sq[3]);
    WAIT_DS_0();                              // LDS reads done before slot reuse
    async_row_load_b128(lds0 + (unsigned)(i & (SLOTS - 1)) * slot_step,
                        (unsigned long long)(src + (size_t)(i + SLOTS) * ROW_BYTES));
    async_row_load_b128(lds0 + (unsigned)((i + 1) & (SLOTS - 1)) * slot_step,
                        (unsigned long long)(src + (size_t)(i + SLOTS + 1) * ROW_BYTES));
  }
  WAIT_ASYNC_0();                             // drain: final SLOTS rows resident
  for (; i < chlen; ++i) {
    float4 v = buf[i & (SLOTS - 1)][tid];
    const float tf = (float)(t0 + i);
    isi_update(v.x, tf, cnt[0], first[0], last[0], sq[0]);
    isi_update(v.y, tf, cnt[1], first[1], last[1], sq[1]);
    isi_update(v.z, tf, cnt[2], first[2], last[2], sq[2]);
    isi_update(v.w, tf, cnt[3], first[3], last[3], sq[3]);
  }

  float4* dst = part + (size_t)c * N_NEURONS + (size_t)tid * 4;
#pragma unroll
  for (int j = 0; j < 4; ++j)
    dst[j] = make_float4(cnt[j], first[j], last[j], sq[j]);
}

// ---------------- stage 2: combine chunks per neuron, per-block partial sums ----------------
__global__ __launch_bounds__(64) void cv_stage2(const float4* __restrict__ part,
                                                const float* __restrict__ target,
                                                float* __restrict__ partials,
                                                int nchunk) {
  const int n = blockIdx.x * 64 + threadIdx.x;   // neuron id
  float cnt = 0.f, first = -1.f, last = -1.f, sq = 0.f;
  for (int c = 0; c < nchunk; ++c) {
    float4 s = part[(size_t)c * N_NEURONS + n];  // coalesced across lanes
    if (s.x > 0.f) {
      if (cnt > 0.f) { float g = s.y - last; sq += g * g; }   // crossing gap
      else first = s.y;
      sq  += s.w;
      last = s.z;
      cnt += s.x;
    }
  }
  const float k    = cnt;
  const float nisi = k - 1.f;
  const float sumg = (k > 0.f) ? (last - first) : 0.f;   // telescoped gap sum (exact)
  const float mean = sumg / fmaxf(nisi, 1.f);
  const float var  = (sq - nisi * mean * mean) / fmaxf(nisi - 1.f, 1.f);
  const float sd   = sqrtf(fmaxf(var, 0.f));
  const bool valid = (k >= 3.f) && (mean > 0.f);
  const float cv   = valid ? (sd / mean) : 0.f;
  const float d    = cv - target[n];

  __shared__ float s_sq[64];
  __shared__ float s_v[64];
  s_sq[threadIdx.x] = valid ? d * d : 0.f;
  s_v[threadIdx.x]  = valid ? 1.f : 0.f;
  __syncthreads();
  if (threadIdx.x == 0) {
    float a = 0.f, b = 0.f;
    for (int t = 0; t < 64; ++t) { a += s_sq[t]; b += s_v[t]; }   // fixed order
    partials[blockIdx.x * 2 + 0] = a;
    partials[blockIdx.x * 2 + 1] = b;
  }
}

// ---------------- stage 3: final deterministic combine ----------------
__global__ void cv_stage3(const float* __restrict__ partials, float* __restrict__ out) {
  if (threadIdx.x == 0 && blockIdx.x == 0) {
    float s = 0.f, v = 0.f;
    for (int i = 0; i < 16; ++i) { s += partials[2 * i]; v += partials[2 * i + 1]; }
    out[0] = (v > 0.f) ? (s / v) : 0.f;
  }
}

// ---------------- launcher ----------------
extern "C" void kernel_launch(void* const* d_in, const int* in_sizes, int n_in,
                              void* d_out, int out_size, void* d_ws, size_t ws_size,
                              hipStream_t stream) {
  const float* spikes = (const float*)d_in[0];   // (B,T,N) fp32 flattened
  const float* target = (const float*)d_in[1];   // (N,) fp32
  const int ttot = in_sizes[0] / in_sizes[1];    // 32768

  // Pick chunk count (grid parallelism) that fits the workspace.
  int nchunk = 512;
  while (nchunk > 16 &&
         (size_t)nchunk * N_NEURONS * sizeof(float4) + 256 > ws_size)
    nchunk >>= 1;
  const int chlen = ttot / nchunk;               // >= 64, multiple of 4

  float4* part     = (float4*)d_ws;
  float*  partials = (float*)((char*)d_ws + (size_t)nchunk * N_NEURONS * sizeof(float4));

  cv_stage1<<<nchunk, T1, 0, stream>>>(spikes, part, chlen);
  cv_stage2<<<N_NEURONS / 64, 64, 0, stream>>>(part, target, partials, nchunk);
  cv_stage3<<<1, 32, 0, stream>>>(partials, (float*)d_out);
}